// TexturedSoftPhongShader_8014408974813
// MI455X (gfx1250) — compile-verified
//
#include <hip/hip_runtime.h>
#include <hip/hip_bf16.h>

typedef float v2f __attribute__((ext_vector_type(2)));
typedef float v8f __attribute__((ext_vector_type(8)));

namespace {
constexpr int KDIM  = 4;     // depth layers per pixel
constexpr int BLOCK = 128;   // 4 waves (wave32)
constexpr int ROWS  = BLOCK * KDIM;   // 512 pixel-layer rows per block
constexpr int YSTR  = 12;    // Y padded 9 -> 12 (K for 3 wmma steps of K=4)

// SH constants (products evaluated in double at compile time)
constexpr float kA0C0 = (float)(3.141592653589793  * 0.28209479177387814);
constexpr float kA1C1 = (float)(3.6275987284684357 * 0.48860251190291987);
constexpr float kA2C2 = (float)(2.221441469079183  * 1.0925484305920792);
constexpr float kD0   = 0.28867513459481287f;
constexpr float SIGMA_INV = 1.0e4f;   // 1/1e-4
constexpr float GAMMA_INV = 1.0e4f;   // 1/1e-4
constexpr float EPSV  = 1.0e-10f;
constexpr float ZFARV = 100.0f, ZNEARV = 1.0f;
}

__global__ __launch_bounds__(BLOCK)
void phong_soft_blend_kernel(const float* __restrict__ texels,
                             const float* __restrict__ bary,
                             const float* __restrict__ zbuf,
                             const float* __restrict__ dists,
                             const float* __restrict__ vnorm,
                             const float* __restrict__ gamma,
                             const int*  __restrict__ p2f,
                             const int*  __restrict__ faces,
                             float* __restrict__ outp,
                             int hw_total)
{
  __shared__ float Ylds[ROWS * YSTR];  // 24 KB: SH basis rows (WMMA A operand)
  __shared__ float Llds[ROWS * 4];     //  8 KB: lighting RGB per row (WMMA D)

  const int tid = threadIdx.x;
  const int n   = blockIdx.y;
  const int pix = blockIdx.x * BLOCK + tid;           // hw index, exact multiple -> no tail
  const long long base = ((long long)n * hw_total + pix) * KDIM;

  // ---- coalesced 128-bit loads of all per-pixel data ----
  const float4 zb = *(const float4*)(zbuf  + base);
  const float4 dd = *(const float4*)(dists + base);
  const int4   pf = *(const int4*)  (p2f   + base);
  const float4 Bq0 = *(const float4*)(bary + base*3);
  const float4 Bq1 = *(const float4*)(bary + base*3 + 4);
  const float4 Bq2 = *(const float4*)(bary + base*3 + 8);
  const float4 Tq0 = *(const float4*)(texels + base*3);
  const float4 Tq1 = *(const float4*)(texels + base*3 + 4);
  const float4 Tq2 = *(const float4*)(texels + base*3 + 8);

  const int   fk[4]    = {pf.x, pf.y, pf.z, pf.w};
  const float bk[4][3] = {{Bq0.x,Bq0.y,Bq0.z},{Bq0.w,Bq1.x,Bq1.y},
                          {Bq1.z,Bq1.w,Bq2.x},{Bq2.y,Bq2.z,Bq2.w}};
  const float tx[4][3] = {{Tq0.x,Tq0.y,Tq0.z},{Tq0.w,Tq1.x,Tq1.y},
                          {Tq1.z,Tq1.w,Tq2.x},{Tq2.y,Tq2.z,Tq2.w}};
  const float zk[4] = {zb.x, zb.y, zb.z, zb.w};
  const float dk[4] = {dd.x, dd.y, dd.z, dd.w};

  // ---- gather + interpolate normals, build SH basis rows in LDS ----
#pragma unroll
  for (int k = 0; k < KDIM; ++k) {
    const bool m = fk[k] >= 0;
    const int fi = m ? fk[k] : 0;
    const int i0 = faces[3*fi+0], i1 = faces[3*fi+1], i2 = faces[3*fi+2];
    const float w0 = bk[k][0], w1 = bk[k][1], w2 = bk[k][2];
    // flip x,y of vertex normals, barycentric interpolation
    float nx = -(w0*vnorm[3*i0+0] + w1*vnorm[3*i1+0] + w2*vnorm[3*i2+0]);
    float ny = -(w0*vnorm[3*i0+1] + w1*vnorm[3*i1+1] + w2*vnorm[3*i2+1]);
    float nz =  (w0*vnorm[3*i0+2] + w1*vnorm[3*i1+2] + w2*vnorm[3*i2+2]);
    if (!m) { nx = 0.f; ny = 0.f; nz = 0.f; }
    float* Y = &Ylds[(tid*KDIM + k) * YSTR];
    Y[0]  = kA0C0;
    Y[1]  = -kA1C1 * ny;
    Y[2]  =  kA1C1 * nz;
    Y[3]  = -kA1C1 * nx;
    Y[4]  =  kA2C2 * nx * ny;
    Y[5]  = -kA2C2 * ny * nz;
    Y[6]  =  kA2C2 * kD0 * (3.f*nz*nz - 1.f);
    Y[7]  = -kA2C2 * nx * nz;
    Y[8]  =  kA2C2 * 0.5f * (nx*nx - ny*ny);
    Y[9]  = 0.f; Y[10] = 0.f; Y[11] = 0.f;   // K padding for wmma step j=2
  }
  __syncthreads();

  // ---- lighting = Y(512x12) @ g(12x16, cols 0..2 live) via V_WMMA_F32_16X16X4_F32 ----
  const int lane = tid & 31;
  const int col  = lane & 15;
  const bool hi  = lane >= 16;                 // lanes 16-31 carry K=2,3 of each step
  const int waveRowBase = (tid >> 5) * (32 * KDIM);

  // B fragments: B[kk][col] = gamma[n][col*9+kk] (+0.8 at kk==0); zero-padded
  const float* g27 = gamma + n * 27;
  v2f bfrag[3];
#pragma unroll
  for (int j = 0; j < 3; ++j) {
    const int kk = 4*j + (hi ? 2 : 0);
    float e0 = 0.f, e1 = 0.f;
    if (col < 3) {
      if (kk   < 9) e0 = g27[col*9 + kk] + (kk == 0 ? 0.8f : 0.f);
      if (kk+1 < 9) e1 = g27[col*9 + kk + 1];
    }
    bfrag[j].x = e0; bfrag[j].y = e1;
  }

#pragma unroll
  for (int t = 0; t < 8; ++t) {                 // 8 tiles of 16 rows per wave
    const int rowbase = waveRowBase + t*16;
    const float* Yrow = &Ylds[(rowbase + col) * YSTR];
    v8f acc = {0.f,0.f,0.f,0.f,0.f,0.f,0.f,0.f};
#pragma unroll
    for (int j = 0; j < 3; ++j) {               // K = 9 padded to 12 = 3 x 4
      const int kk = 4*j + (hi ? 2 : 0);
      v2f a; a.x = Yrow[kk]; a.y = Yrow[kk + 1];
      acc = __builtin_amdgcn_wmma_f32_16x16x4_f32(false, a, false, bfrag[j],
                                                  (short)0, acc, false, false);
    }
    // D layout: VGPR r -> (M=r, N=lane) for lanes 0-15, (M=8+r, N=lane-16) for 16-31
    if (col < 3) {
      const int mbase = rowbase + (hi ? 8 : 0);
#pragma unroll
      for (int r = 0; r < 8; ++r)
        Llds[(mbase + r)*4 + col] = acc[r];
    }
  }
  __syncthreads();

  // ---- softmax RGB blend across K layers ----
  float prob[4], zinv[4];
  float zmax = EPSV;
#pragma unroll
  for (int k = 0; k < KDIM; ++k) {
    const float m = fk[k] >= 0 ? 1.f : 0.f;
    prob[k] = m / (1.f + expf(dk[k] * SIGMA_INV));          // sigmoid(-d/sigma)*mask
    zinv[k] = (ZFARV - zk[k]) * (1.f/(ZFARV - ZNEARV)) * m;
    zmax = fmaxf(zmax, zinv[k]);
  }
  float wnum[4], wsum = 0.f, keep = 1.f;
#pragma unroll
  for (int k = 0; k < KDIM; ++k) {
    wnum[k] = prob[k] * expf((zinv[k] - zmax) * GAMMA_INV);
    wsum += wnum[k];
    keep *= (1.f - prob[k]);
  }
  const float delta = fmaxf(expf((EPSV - zmax) * GAMMA_INV), EPSV);
  const float inv   = 1.f / (wsum + delta);
  float rgb[3];
#pragma unroll
  for (int c = 0; c < 3; ++c) {
    float s = delta;                                        // + delta * BG(=1.0)
#pragma unroll
    for (int k = 0; k < KDIM; ++k)
      s += wnum[k] * (Llds[(tid*KDIM + k)*4 + c] * tx[k][c]);
    rgb[c] = s * inv;
  }
  float4 o; o.x = rgb[0]; o.y = rgb[1]; o.z = rgb[2]; o.w = 1.f - keep;
  *(float4*)(outp + ((long long)n * hw_total + pix) * 4) = o;
}

extern "C" void kernel_launch(void* const* d_in, const int* in_sizes, int n_in,
                              void* d_out, int out_size, void* d_ws, size_t ws_size,
                              hipStream_t stream) {
  const float* texels = (const float*)d_in[0];
  const float* bary   = (const float*)d_in[1];
  const float* zbuf   = (const float*)d_in[2];
  const float* dists  = (const float*)d_in[3];
  // d_in[4] = verts: unused (specular term is zero in the reference)
  const float* vnorm  = (const float*)d_in[5];
  const float* gamma  = (const float*)d_in[6];
  const int*   p2f    = (const int*)d_in[7];
  const int*   faces  = (const int*)d_in[8];
  float* outp = (float*)d_out;

  const int N  = in_sizes[6] / 27;              // gamma is (N,27)
  const int hw = in_sizes[2] / (N * KDIM);      // zbuf is (N,H,W,K)
  dim3 grid(hw / BLOCK, N);
  phong_soft_blend_kernel<<<grid, BLOCK, 0, stream>>>(
      texels, bary, zbuf, dists, vnorm, gamma, p2f, faces, outp, hw);
}